// DefaultFilterer_53515292508478
// MI455X (gfx1250) — compile-verified
//
#include <hip/hip_runtime.h>

// ids are in [0, 100000) -> 3125 x 32-bit words per bitmap; pad to 3136 (16B multiple)
#define NWORDS 3125
#define NWPAD  3136
#define SCAN_BLOCKS 1024
#define SCAN_THREADS 256

#define AS1 __attribute__((address_space(1)))
#define AS3 __attribute__((address_space(3)))

typedef int v4i __attribute__((vector_size(16)));

#if defined(__gfx1250__) && __has_builtin(__builtin_amdgcn_global_load_async_to_lds_b128) && __has_builtin(__builtin_amdgcn_s_wait_asynccnt)
#define HAVE_GFX1250_ASYNC 1
#else
#define HAVE_GFX1250_ASYNC 0
#endif

// ---------------- kernel 1: zero the three bitmaps in workspace ----------------
__global__ void df_zero_bitmaps(unsigned* __restrict__ ws) {
    int i = blockIdx.x * blockDim.x + threadIdx.x;
    if (i < 3 * NWPAD) ws[i] = 0u;
}

// ---------------- kernel 2: mark negative heads & relations ----------------
__global__ void df_mark_negatives(const int* __restrict__ neg, int B,
                                  unsigned* __restrict__ ws) {
    int b = blockIdx.x * blockDim.x + threadIdx.x;
    if (b >= B) return;
    unsigned h = (unsigned)neg[b * 3 + 0];
    unsigned r = (unsigned)neg[b * 3 + 1];
    atomicOr(&ws[h >> 5], 1u << (h & 31u));                 // head bitmap
    atomicOr(&ws[NWPAD + (r >> 5)], 1u << (r & 31u));       // relation bitmap
}

// ---------------- kernel 3: stream 16M triples, collect surviving tails ----------------
__global__ __launch_bounds__(SCAN_THREADS) void df_scan_triples(
    const int* __restrict__ mt, int N,
    const unsigned* __restrict__ ws,        // head bitmap @0, rel bitmap @NWPAD (contiguous)
    unsigned* __restrict__ tail_global) {   // tail bitmap in global ws
    __shared__ unsigned lds_hr[2 * NWPAD];  // head bits then rel bits
    __shared__ unsigned lds_tail[NWPAD];

    // zero per-block tail bitmap
    for (int i = threadIdx.x; i < NWPAD; i += blockDim.x) lds_tail[i] = 0u;

    // stage head+rel bitmaps (25088 B) global -> LDS
#if HAVE_GFX1250_ASYNC
    for (int i = threadIdx.x; i < (2 * NWPAD) / 4; i += blockDim.x) {
        __builtin_amdgcn_global_load_async_to_lds_b128(
            (AS1 v4i*)(ws + 4 * i), (AS3 v4i*)(&lds_hr[4 * i]), 0, 0);
    }
    __builtin_amdgcn_s_wait_asynccnt(0);
#else
    for (int i = threadIdx.x; i < (2 * NWPAD) / 4; i += blockDim.x) {
        ((uint4*)lds_hr)[i] = ((const uint4*)ws)[i];
    }
#endif
    __syncthreads();

    const unsigned* lds_head = lds_hr;
    const unsigned* lds_rel  = lds_hr + NWPAD;

    int units  = (N + 3) >> 2;                       // 4 triples (48 B) per unit
    int stride = gridDim.x * blockDim.x;

    for (int u = blockIdx.x * blockDim.x + threadIdx.x; u < units; u += stride) {
        size_t t0 = (size_t)u * 4;
        // prefetch the next grid-stride chunk into the cache hierarchy
        if (u + stride < units) {
            __builtin_prefetch(mt + (size_t)(u + stride) * 12, 0, 1);
        }
        if (t0 + 4 <= (size_t)N) {
            const uint4* p = (const uint4*)(mt + t0 * 3);
            uint4 a = p[0], b = p[1], c = p[2];
            unsigned hs[4] = {a.x, a.w, b.z, c.y};
            unsigned rs[4] = {a.y, b.x, b.w, c.z};
            unsigned ts[4] = {a.z, b.y, c.x, c.w};
#pragma unroll
            for (int k = 0; k < 4; ++k) {
                unsigned hv = (lds_head[hs[k] >> 5] >> (hs[k] & 31u)) & 1u;
                unsigned rv = (lds_rel [rs[k] >> 5] >> (rs[k] & 31u)) & 1u;
                if (hv & rv) atomicOr((unsigned*)&lds_tail[ts[k] >> 5], 1u << (ts[k] & 31u));
            }
        } else {
            for (size_t t = t0; t < (size_t)N; ++t) {
                unsigned h = (unsigned)mt[t * 3 + 0];
                unsigned r = (unsigned)mt[t * 3 + 1];
                unsigned tt = (unsigned)mt[t * 3 + 2];
                unsigned hv = (lds_head[h >> 5] >> (h & 31u)) & 1u;
                unsigned rv = (lds_rel [r >> 5] >> (r & 31u)) & 1u;
                if (hv & rv) atomicOr((unsigned*)&lds_tail[tt >> 5], 1u << (tt & 31u));
            }
        }
    }
    __syncthreads();

    // merge this block's tail bits into the global tail bitmap (skip empty words)
    for (int i = threadIdx.x; i < NWORDS; i += blockDim.x) {
        unsigned w = lds_tail[i];
        if (w) atomicOr(&tail_global[i], w);
    }
}

// ---------------- kernel 4: filter negatives, emit float outputs ----------------
__global__ void df_finalize(const int* __restrict__ neg, int B,
                            const unsigned* __restrict__ tail_bits,
                            float* __restrict__ out) {
    int b = blockIdx.x * blockDim.x + threadIdx.x;
    if (b >= B) return;
    int h = neg[b * 3 + 0];
    int r = neg[b * 3 + 1];
    int t = neg[b * 3 + 2];
    unsigned hit = (tail_bits[(unsigned)t >> 5] >> ((unsigned)t & 31u)) & 1u;
    // keep = !final_filter ; filtered rows are -1 per reference
    out[b * 3 + 0] = hit ? -1.0f : (float)h;
    out[b * 3 + 1] = hit ? -1.0f : (float)r;
    out[b * 3 + 2] = hit ? -1.0f : (float)t;
    out[3 * B + b] = hit ? 0.0f : 1.0f;   // keep_mask
}

extern "C" void kernel_launch(void* const* d_in, const int* in_sizes, int n_in,
                              void* d_out, int out_size, void* d_ws, size_t ws_size,
                              hipStream_t stream) {
    const int* neg = (const int*)d_in[0];        // [B,3] int32
    const int* mt  = (const int*)d_in[1];        // [N,3] int32
    int B = in_sizes[0] / 3;
    int N = in_sizes[1] / 3;

    unsigned* ws        = (unsigned*)d_ws;       // head @0, rel @NWPAD, tail @2*NWPAD
    unsigned* tail_bits = ws + 2 * NWPAD;
    float* out = (float*)d_out;

    df_zero_bitmaps<<<(3 * NWPAD + 255) / 256, 256, 0, stream>>>(ws);
    df_mark_negatives<<<(B + 255) / 256, 256, 0, stream>>>(neg, B, ws);
    df_scan_triples<<<SCAN_BLOCKS, SCAN_THREADS, 0, stream>>>(mt, N, ws, tail_bits);
    df_finalize<<<(B + 255) / 256, 256, 0, stream>>>(neg, B, tail_bits, out);
}